// MultiHeadSimpleGraphAttention_54211077210471
// MI455X (gfx1250) — compile-verified
//
#include <hip/hip_runtime.h>
#include <math.h>

// MultiHeadSimpleGraphAttention for MI455X (gfx1250, wave32).
// Memory-bound (~24 GF vs ~1 GB traffic @ 23.3 TB/s) -> use fp32 WMMA
// (V_WMMA_F32_16X16X4_F32) to keep full precision; nq/nk/nv (61 MB) stay
// L2-resident (192 MB) so per-edge gathers are cheap.

#define NNODES 10000
#define NEDGES 160000
#define DEMB   64
#define NHEADS 8
#define HDIM   512   // NHEADS * DEMB

typedef __attribute__((ext_vector_type(2))) float v2f;
typedef __attribute__((ext_vector_type(8))) float v8f;

// ---------------------------------------------------------------------------
// Generic fp32 WMMA GEMM:  C[M,N] = A[M,K] @ W[N,K]^T + bias[N]
// Requirements: M % 16 == 0, N % 64 == 0, K % 4 == 0 (K is 64 or 512 here).
// One wave computes a 16x64 strip (4 WMMA tiles) -> 4x reuse of each A frag.
// Software-pipelined: fragments for step k+4 are fetched before the four
// WMMAs of step k issue, so global_load latency overlaps matrix work instead
// of forcing s_wait_loadcnt 0 in front of every v_wmma.
// A-frag layout (f32 16x4): lane<16 -> M=lane, K={k0,k0+1}; lane>=16 -> K={k0+2,k0+3}.
// B-frag layout (f32 4x16): lane<16 -> N=lane, rows {k0,k0+1}; lane>=16 -> rows {k0+2,k0+3}.
// C/D layout: VGPR r, lane<16 -> [M=r][N=lane]; lane>=16 -> [M=r+8][N=lane-16].
// ---------------------------------------------------------------------------
__global__ void gemm_bias_f32_wmma(const float* __restrict__ A,
                                   const float* __restrict__ W,
                                   const float* __restrict__ bias,
                                   float* __restrict__ C,
                                   int M, int N, int K)
{
    const int wid  = (blockIdx.x * blockDim.x + threadIdx.x) >> 5;
    const int lane = threadIdx.x & 31;
    const int tiles_n = N >> 6;              // 64-wide column groups per row-tile
    const int tiles_m = M >> 4;
    const int tm = wid / tiles_n;
    const int tn = wid - tm * tiles_n;
    if (tm >= tiles_m) return;               // wave-uniform: EXEC all-ones for WMMA

    const int row0 = tm << 4;
    const int col0 = tn << 6;
    const int hi   = lane >> 4;              // K-half select (0 or 1)
    const int lr   = lane & 15;              // M index (A) / N index (B)

    const float* arow = A + (size_t)(row0 + lr) * K + (hi << 1);
    const float* w0   = W + (size_t)(col0 +  0 + lr) * K + (hi << 1);
    const float* w1   = W + (size_t)(col0 + 16 + lr) * K + (hi << 1);
    const float* w2   = W + (size_t)(col0 + 32 + lr) * K + (hi << 1);
    const float* w3   = W + (size_t)(col0 + 48 + lr) * K + (hi << 1);

    v8f acc0 = {}, acc1 = {}, acc2 = {}, acc3 = {};

    // prologue: fragments for k0 = 0
    v2f a  = *(const v2f*)(arow);
    v2f b0 = *(const v2f*)(w0);
    v2f b1 = *(const v2f*)(w1);
    v2f b2 = *(const v2f*)(w2);
    v2f b3 = *(const v2f*)(w3);

    for (int k0 = 0; k0 < K; k0 += 4) {
        // prefetch next step's fragments (clamped on last iter: harmless reload,
        // keeps the loads unconditional so the scheduler can hoist them)
        const int kn = (k0 + 4 < K) ? (k0 + 4) : k0;
        v2f an  = *(const v2f*)(arow + kn);
        v2f bn0 = *(const v2f*)(w0 + kn);
        v2f bn1 = *(const v2f*)(w1 + kn);
        v2f bn2 = *(const v2f*)(w2 + kn);
        v2f bn3 = *(const v2f*)(w3 + kn);

        acc0 = __builtin_amdgcn_wmma_f32_16x16x4_f32(false, a, false, b0, (short)0, acc0, false, false);
        acc1 = __builtin_amdgcn_wmma_f32_16x16x4_f32(false, a, false, b1, (short)0, acc1, false, false);
        acc2 = __builtin_amdgcn_wmma_f32_16x16x4_f32(false, a, false, b2, (short)0, acc2, false, false);
        acc3 = __builtin_amdgcn_wmma_f32_16x16x4_f32(false, a, false, b3, (short)0, acc3, false, false);

        a = an; b0 = bn0; b1 = bn1; b2 = bn2; b3 = bn3;
    }

    const float bv0 = bias[col0 +  0 + lr];
    const float bv1 = bias[col0 + 16 + lr];
    const float bv2 = bias[col0 + 32 + lr];
    const float bv3 = bias[col0 + 48 + lr];
#pragma unroll
    for (int r = 0; r < 8; ++r) {
        const int m = row0 + r + (hi << 3);
        float* crow = C + (size_t)m * N + col0 + lr;
        crow[ 0] = acc0[r] + bv0;
        crow[16] = acc1[r] + bv1;
        crow[32] = acc2[r] + bv2;
        crow[48] = acc3[r] + bv3;
    }
}

// ---------------------------------------------------------------------------
__device__ __forceinline__ float atomicMaxF(float* addr, float val)
{
    unsigned int* a = (unsigned int*)addr;
    unsigned int old = *a;
    while (__uint_as_float(old) < val) {
        unsigned int assumed = old;
        old = atomicCAS(a, assumed, __float_as_uint(val));
        if (old == assumed) break;
    }
    return __uint_as_float(old);
}

// init m = -inf, denom = 0, node_agg = 0
__global__ void init_buffers(float* __restrict__ mvec,
                             float* __restrict__ denom,
                             float* __restrict__ nagg)
{
    const int idx = blockIdx.x * blockDim.x + threadIdx.x;
    if (idx < NNODES * NHEADS) {
        mvec[idx]  = -__builtin_inff();
        denom[idx] = 0.0f;
    }
    if (idx < NNODES * HDIM) nagg[idx] = 0.0f;
}

// score[e,h] = dot(nq[src], nk[dst]+ek[e]) / 8 ;  m[src,h] = max(...)
// One wave per edge; 4 lanes per head, 16 contiguous dims per lane.
__global__ void score_kernel(const float* __restrict__ nq,
                             const float* __restrict__ nk,
                             const float* __restrict__ ek,
                             const int* __restrict__ src,
                             const int* __restrict__ dst,
                             float* __restrict__ score,
                             float* __restrict__ mvec)
{
    const int wid  = (blockIdx.x * blockDim.x + threadIdx.x) >> 5;
    const int lane = threadIdx.x & 31;
    if (wid >= NEDGES) return;
    const int e = wid;
    const int s = src[e];
    const int d = dst[e];
    const int h = lane >> 2;
    const int q = lane & 3;
    const int base = h * DEMB + q * 16;

    const float4* pq = (const float4*)(nq + (size_t)s * HDIM + base);
    const float4* pk = (const float4*)(nk + (size_t)d * HDIM + base);
    const float4* pe = (const float4*)(ek + (size_t)e * HDIM + base);

    float acc = 0.0f;
#pragma unroll
    for (int i = 0; i < 4; ++i) {
        float4 a = pq[i], b = pk[i], c = pe[i];
        acc += a.x * (b.x + c.x) + a.y * (b.y + c.y)
             + a.z * (b.z + c.z) + a.w * (b.w + c.w);
    }
    acc += __shfl_xor(acc, 1, 32);
    acc += __shfl_xor(acc, 2, 32);
    acc *= 0.125f;                         // 1/sqrt(64)
    if (q == 0) {
        score[(size_t)e * NHEADS + h] = acc;
        atomicMaxF(&mvec[s * NHEADS + h], acc);
    }
}

// p = exp(score - m[src]); denom[src,h] += p; node_agg[src,h,:] += p * nv[dst,h,:]
__global__ void softmax_scatter(const float* __restrict__ score,
                                const float* __restrict__ mvec,
                                const float* __restrict__ nv,
                                const int* __restrict__ src,
                                const int* __restrict__ dst,
                                float* __restrict__ denom,
                                float* __restrict__ nagg)
{
    const int wid  = (blockIdx.x * blockDim.x + threadIdx.x) >> 5;
    const int lane = threadIdx.x & 31;
    if (wid >= NEDGES) return;
    const int e = wid;
    const int s = src[e];
    const int d = dst[e];
    const int h = lane >> 2;
    const int q = lane & 3;

    const float p = expf(score[(size_t)e * NHEADS + h] - mvec[s * NHEADS + h]);
    if (q == 0) atomicAdd(&denom[s * NHEADS + h], p);

    const int base = h * DEMB + q * 16;
    const float4* pv = (const float4*)(nv + (size_t)d * HDIM + base);
    float* pa = nagg + (size_t)s * HDIM + base;
#pragma unroll
    for (int i = 0; i < 4; ++i) {
        float4 v = pv[i];
        atomicAdd(pa + 4 * i + 0, p * v.x);
        atomicAdd(pa + 4 * i + 1, p * v.y);
        atomicAdd(pa + 4 * i + 2, p * v.z);
        atomicAdd(pa + 4 * i + 3, p * v.w);
    }
}

// node_agg[n,h,:] /= denom[n,h]  (guard nodes with no out-edges)
__global__ void normalize_agg(float* __restrict__ nagg,
                              const float* __restrict__ denom)
{
    const int idx = blockIdx.x * blockDim.x + threadIdx.x;
    if (idx >= NNODES * HDIM) return;
    const int n = idx >> 9;
    const int h = (idx >> 6) & (NHEADS - 1);
    const float dn = denom[n * NHEADS + h];
    nagg[idx] = (dn > 0.0f) ? nagg[idx] / dn : 0.0f;
}

// ---------------------------------------------------------------------------
extern "C" void kernel_launch(void* const* d_in, const int* in_sizes, int n_in,
                              void* d_out, int out_size, void* d_ws, size_t ws_size,
                              hipStream_t stream)
{
    const float* x    = (const float*)d_in[0];
    const int*   eidx = (const int*)  d_in[1];
    const float* ef   = (const float*)d_in[2];
    const float* Wq   = (const float*)d_in[3];
    const float* bq   = (const float*)d_in[4];
    const float* Wk   = (const float*)d_in[5];
    const float* bk   = (const float*)d_in[6];
    const float* Wv   = (const float*)d_in[7];
    const float* bv   = (const float*)d_in[8];
    const float* Wek  = (const float*)d_in[9];
    const float* bek  = (const float*)d_in[10];
    const float* Wl   = (const float*)d_in[11];
    const float* bl   = (const float*)d_in[12];
    const float* Wel  = (const float*)d_in[13];
    const float* bel  = (const float*)d_in[14];

    const int* src = eidx;            // edge_index[0] = reduce/segment index
    const int* dst = eidx + NEDGES;   // edge_index[1]

    // workspace carve-up (floats): ~415 MB total
    float* nq    = (float*)d_ws;
    float* nk    = nq    + (size_t)NNODES * HDIM;      //  5,120,000
    float* nv    = nk    + (size_t)NNODES * HDIM;
    float* ekb   = nv    + (size_t)NNODES * HDIM;      // 81,920,000
    float* score = ekb   + (size_t)NEDGES * HDIM;      //  1,280,000
    float* mvec  = score + (size_t)NEDGES * NHEADS;    //     80,000
    float* denom = mvec  + (size_t)NNODES * NHEADS;    //     80,000
    float* nagg  = denom + (size_t)NNODES * NHEADS;    //  5,120,000

    float* out_node = (float*)d_out;                       // [10000 x 64]
    float* out_edge = out_node + (size_t)NNODES * DEMB;    // [160000 x 64]

    const int BLK = 256;   // 8 waves/block
    auto gemm_blocks = [](int M, int N) {
        int waves = (M >> 4) * (N >> 6);
        return (waves * 32 + 255) / 256;
    };

    // 1) init reduction buffers
    init_buffers<<<(NNODES * HDIM + BLK - 1) / BLK, BLK, 0, stream>>>(mvec, denom, nagg);

    // 2-5) input projections (fp32 WMMA GEMMs)
    gemm_bias_f32_wmma<<<gemm_blocks(NNODES, HDIM), BLK, 0, stream>>>(x,  Wq,  bq,  nq,  NNODES, HDIM, DEMB);
    gemm_bias_f32_wmma<<<gemm_blocks(NNODES, HDIM), BLK, 0, stream>>>(x,  Wk,  bk,  nk,  NNODES, HDIM, DEMB);
    gemm_bias_f32_wmma<<<gemm_blocks(NNODES, HDIM), BLK, 0, stream>>>(x,  Wv,  bv,  nv,  NNODES, HDIM, DEMB);
    gemm_bias_f32_wmma<<<gemm_blocks(NEDGES, HDIM), BLK, 0, stream>>>(ef, Wek, bek, ekb, NEDGES, HDIM, DEMB);

    // 6) per-edge scores + segment max
    score_kernel<<<(NEDGES * 32 + BLK - 1) / BLK, BLK, 0, stream>>>(nq, nk, ekb, src, dst, score, mvec);

    // 7) softmax numerator + weighted-V scatter
    softmax_scatter<<<(NEDGES * 32 + BLK - 1) / BLK, BLK, 0, stream>>>(score, mvec, nv, src, dst, denom, nagg);

    // 8) divide by segment sum
    normalize_agg<<<(NNODES * HDIM + BLK - 1) / BLK, BLK, 0, stream>>>(nagg, denom);

    // 9-10) output projections straight into d_out
    gemm_bias_f32_wmma<<<gemm_blocks(NNODES, DEMB), BLK, 0, stream>>>(nagg, Wl,  bl,  out_node, NNODES, DEMB, HDIM);
    gemm_bias_f32_wmma<<<gemm_blocks(NEDGES, DEMB), BLK, 0, stream>>>(ekb,  Wel, bel, out_edge, NEDGES, DEMB, HDIM);
}